// PredictionMemorySystem_70068096467340
// MI455X (gfx1250) — compile-verified
//
#include <hip/hip_runtime.h>
#include <hip/hip_bf16.h>

// Problem constants (fixed by the reference)
constexpr int M = 100000;    // MEMORY_SIZE
constexpr int D = 1024;      // FEATURE_DIM
constexpr int B = 4096;      // BATCH
constexpr long long MD  = (long long)M * D;        // 102,400,000 floats per tensor
constexpr long long MD4 = MD / 4;                  // 25,600,000 float4 per tensor

typedef __attribute__((ext_vector_type(2))) float v2f;
typedef __attribute__((ext_vector_type(8))) float v8f;

// ---------------------------------------------------------------------------
// Streaming circular-buffer merge for features / predictions.
// grid.y selects tensor (0 = features, 1 = predictions).
// i indexes float4 elements; 256 float4 == one 1024-float row, and the block
// size is 256, so every wave (and every block-iteration) lies entirely inside
// one row -> the window test is wave-uniform (scalar branch).
// ---------------------------------------------------------------------------
__global__ void pm_copy_rows_kernel(const float4* __restrict__ feat_in,
                                    const float4* __restrict__ pred_in,
                                    const float4* __restrict__ feat_mem,
                                    const float4* __restrict__ pred_mem,
                                    const int* __restrict__ p_idx,
                                    float4* __restrict__ out)
{
    const int which = blockIdx.y;
    const float4* __restrict__ in  = which ? pred_in  : feat_in;
    const float4* __restrict__ mem = which ? pred_mem : feat_mem;
    float4* __restrict__ o = out + (long long)which * MD4;

    const int mem_index = *p_idx;

    long long i = (long long)blockIdx.x * blockDim.x + threadIdx.x;
    const long long stride = (long long)gridDim.x * blockDim.x;

    for (; i < MD4; i += stride) {
        // Prefetch next iteration's streaming (memory) source into cache.
        // Speculative: translation failures past the end are silently dropped.
        __builtin_prefetch(mem + i + stride, 0, 3);

        const int row = (int)(i >> 8);            // 256 float4 per row
        int off = row - mem_index;
        if (off < 0) off += M;                    // (row - mem_index) mod M

        float4 v;
        if (off < B) {
            v = in[((long long)off << 8) + (i & 255)];
        } else {
            v = mem[i];
        }
        o[i] = v;
    }
}

// ---------------------------------------------------------------------------
// new_conf merge + utilization scalar.
// ---------------------------------------------------------------------------
__global__ void pm_conf_kernel(const float* __restrict__ conf_in,
                               const float* __restrict__ conf_mem,
                               const int* __restrict__ p_idx,
                               const int* __restrict__ p_full,
                               float* __restrict__ out_conf,
                               float* __restrict__ out_scalars)
{
    const int mem_index = *p_idx;
    const int i = blockIdx.x * blockDim.x + threadIdx.x;
    if (i < M) {
        int off = i - mem_index;
        if (off < 0) off += M;
        out_conf[i] = (off < B) ? conf_in[off] : conf_mem[i];
    }
    if (i == 0) {
        const int  new_index = (mem_index + B) % M;
        const bool wrapped   = (mem_index + B) >= M;
        const bool full      = (*p_full != 0) || wrapped;
        // out_scalars[1] = utilization
        out_scalars[1] = full ? 1.0f : (float)new_index / (float)M;
    }
}

// ---------------------------------------------------------------------------
// conf_mean via V_WMMA_F32_16X16X4_F32.
// A = ones(16x4)  =>  D[m][n] = sum_k B[k][n] + C[m][n]   (mult by 1.0 exact).
// Each iteration folds 64 confidences (2 per lane) into the 16 column sums
// held in the accumulator; after the loop, c[0] on lanes 0..15 holds the
// D-row-0 column sums (per the 16x16 f32 C/D layout). A 16-lane shuffle
// reduction yields the total. Single wave, EXEC all ones (WMMA requirement).
// ---------------------------------------------------------------------------
__global__ __launch_bounds__(32)
void pm_conf_mean_wmma_kernel(const float* __restrict__ conf,
                              float* __restrict__ out_mean)
{
    const int lane = threadIdx.x;          // 0..31, wave32

    v2f a; a.x = 1.0f; a.y = 1.0f;         // A = all ones (16x4 f32, 2 VGPRs)
    v8f c = {};                            // C/D accumulator (16x16 f32, 8 VGPRs)

    for (int base = 0; base < B; base += 64) {
        v2f b;
        b.x = conf[base + lane];           // 64 values per WMMA: any bijection
        b.y = conf[base + 32 + lane];      // onto B's (k,n) slots sums correctly
        c = __builtin_amdgcn_wmma_f32_16x16x4_f32(
                /*neg_a=*/false, a, /*neg_b=*/false, b,
                /*c_mod=*/(short)0, c, /*reuse_a=*/false, /*reuse_b=*/false);
    }

    // c[0]: lanes 0..15 = D[0][n] (column sums); lanes 16..31 = D[8][n] (dup).
    float v = c[0];
    for (int off = 8; off > 0; off >>= 1)
        v += __shfl_down(v, off, 16);

    if (lane == 0) *out_mean = v / (float)B;
}

// ---------------------------------------------------------------------------
// Launch
// Output layout (reference return order, flat):
//   [0, MD)          new_feat
//   [MD, 2*MD)       new_pred
//   [2*MD, 2*MD+M)   new_conf
//   [2*MD+M]         conf_mean
//   [2*MD+M+1]       utilization
// ---------------------------------------------------------------------------
extern "C" void kernel_launch(void* const* d_in, const int* in_sizes, int n_in,
                              void* d_out, int out_size, void* d_ws, size_t ws_size,
                              hipStream_t stream)
{
    const float* features    = (const float*)d_in[0];
    const float* predictions = (const float*)d_in[1];
    const float* confidences = (const float*)d_in[2];
    const float* mem_feat    = (const float*)d_in[3];
    const float* mem_pred    = (const float*)d_in[4];
    const float* mem_conf    = (const float*)d_in[5];
    const int*   p_idx       = (const int*)d_in[6];
    const int*   p_full      = (const int*)d_in[7];

    float* out         = (float*)d_out;
    float* out_conf    = out + 2 * MD;
    float* out_scalars = out + 2 * MD + M;   // [0]=conf_mean, [1]=utilization

    // 25000 blocks x 256 threads x 2 tensors; each thread does 4 float4 per
    // tensor via grid-stride (stride = 6.4M float4, MD4/stride = 4 iters).
    dim3 cgrid(25000, 2, 1);
    pm_copy_rows_kernel<<<cgrid, 256, 0, stream>>>(
        (const float4*)features, (const float4*)predictions,
        (const float4*)mem_feat, (const float4*)mem_pred,
        p_idx, (float4*)out);

    pm_conf_kernel<<<(M + 255) / 256, 256, 0, stream>>>(
        confidences, mem_conf, p_idx, p_full, out_conf, out_scalars);

    pm_conf_mean_wmma_kernel<<<1, 32, 0, stream>>>(
        confidences, out_scalars + 0);
}